// Critic_16097537425468
// MI455X (gfx1250) — compile-verified
//
#include <hip/hip_runtime.h>

typedef __attribute__((ext_vector_type(2))) float v2f;
typedef __attribute__((ext_vector_type(8))) float v8f;

#define B_SZ 4096
#define L_SZ 200
#define GD 256
#define LD 128
#define AD 64
#define HID 32

__device__ __forceinline__ v8f wmma_f32(v2f a, v2f b, v8f c) {
  // D = A(16x4) * B(4x16) + C(16x16), all fp32, RNE
  return __builtin_amdgcn_wmma_f32_16x16x4_f32(false, a, false, b, (short)0, c,
                                               false, false);
}

__device__ __forceinline__ float leaky(float x) { return x > 0.f ? x : 0.01f * x; }

// v[d] = sum_h att_w[32+h] * U_w[h,d];  k0 = att_b + sum_h att_w[32+h]*U_b[h]
__global__ void prep_kernel(const float* __restrict__ att_w,
                            const float* __restrict__ att_b,
                            const float* __restrict__ Uw,
                            const float* __restrict__ Ub,
                            float* __restrict__ v, float* __restrict__ k0) {
  int d = threadIdx.x;  // 128 threads
  float acc = 0.f;
  for (int h = 0; h < HID; ++h) acc += att_w[HID + h] * Uw[h * LD + d];
  v[d] = acc;
  if (d == 0) {
    float k = att_b[0];
    for (int h = 0; h < HID; ++h) k += att_w[HID + h] * Ub[h];
    k0[0] = k;
  }
}

// wg = global_states @ W_w.T + W_b   (4096x256 -> 4096x32) via fp32 WMMA
__global__ void wg_kernel(const float* __restrict__ gs,
                          const float* __restrict__ Ww,
                          const float* __restrict__ Wb, float* __restrict__ wg) {
  int lane = threadIdx.x & 31;
  int wave = threadIdx.x >> 5;
  int r0 = (blockIdx.x * 4 + wave) * 16;
  int m = lane & 15, half = lane >> 4;
  v8f acc0 = {}, acc1 = {};
  const float* arow = gs + (size_t)(r0 + m) * GD;
  const float* b0row = Ww + (size_t)m * GD;          // B[k][n]=Ww[n][k], tile n=0..15
  const float* b1row = Ww + (size_t)(16 + m) * GD;   // tile n=16..31
  for (int kb = 0; kb < GD; kb += 4) {
    int k = kb + 2 * half;
    v2f a;  a.x = arow[k];  a.y = arow[k + 1];
    v2f b0; b0.x = b0row[k]; b0.y = b0row[k + 1];
    v2f b1; b1.x = b1row[k]; b1.y = b1row[k + 1];
    acc0 = wmma_f32(a, b0, acc0);
    acc1 = wmma_f32(a, b1, acc1);
  }
  float bias0 = Wb[m], bias1 = Wb[16 + m];
#pragma unroll
  for (int g = 0; g < 8; ++g) {
    int row = r0 + g + 8 * half;
    wg[row * HID + m]      = acc0[g] + bias0;
    wg[row * HID + 16 + m] = acc1[g] + bias1;
  }
}

// per-row score constants: cb[b] = att_w1.wg + k0 ;  s0[b] = leaky((w1+w2).wg + att_b)
__global__ void sconst_kernel(const float* __restrict__ wg,
                              const float* __restrict__ att_w,
                              const float* __restrict__ att_b,
                              const float* __restrict__ k0,
                              float* __restrict__ cb, float* __restrict__ s0) {
  int b = blockIdx.x * blockDim.x + threadIdx.x;
  if (b >= B_SZ) return;
  float c = 0.f, t = 0.f;
  for (int h = 0; h < HID; ++h) {
    float g = wg[b * HID + h];
    float w1 = att_w[h], w2 = att_w[HID + h];
    c += w1 * g;
    t += (w1 + w2) * g;
  }
  cb[b] = c + k0[0];
  s0[b] = leaky(t + att_b[0]);
}

// Fused attention: stream local_states once; produce sa[b] = [relu(g_part), relu(l_part), actions]
__global__ void attn_kernel(const float* __restrict__ locals,
                            const float* __restrict__ v,
                            const float* __restrict__ cb,
                            const float* __restrict__ s0,
                            const float* __restrict__ wg,
                            const float* __restrict__ Uw,
                            const float* __restrict__ Ub,
                            const float* __restrict__ actions,
                            float* __restrict__ sa) {
  __shared__ __align__(16) float sh_m[8][LD];
  __shared__ float sh_s[8];
  __shared__ float sh_mm[LD];
  __shared__ float sh_sc[2];  // s1_total, 1/S
  int b = blockIdx.x;
  int tid = threadIdx.x;
  int lane = tid & 31, wave = tid >> 5;
  const float4 vreg = ((const float4*)v)[lane];
  const float c = cb[b];
  const float4* base = (const float4*)(locals + (size_t)b * L_SZ * LD);
  float4 macc = {0.f, 0.f, 0.f, 0.f};
  float s1 = 0.f;
  for (int l = wave; l < L_SZ; l += 8) {
    float4 x = base[l * 32 + lane];
    float d = x.x * vreg.x + x.y * vreg.y + x.z * vreg.z + x.w * vreg.w;
#pragma unroll
    for (int off = 16; off; off >>= 1) d += __shfl_xor(d, off);
    float sc = leaky(c + d);
    s1 += sc;
    macc.x += sc * x.x; macc.y += sc * x.y; macc.z += sc * x.z; macc.w += sc * x.w;
  }
  *(float4*)&sh_m[wave][lane * 4] = macc;
  if (lane == 0) sh_s[wave] = s1;
  __syncthreads();
  if (tid < LD) {
    float t = 0.f;
    for (int w = 0; w < 8; ++w) t += sh_m[w][tid];
    sh_mm[tid] = t;
  }
  if (tid == 0) {
    float t = 0.f;
    for (int w = 0; w < 8; ++w) t += sh_s[w];
    sh_sc[0] = t;
    sh_sc[1] = 1.f / (s0[b] + t);
  }
  __syncthreads();
  float invS = sh_sc[1];
  if (tid < HID) {
    float gp = s0[b] * invS * wg[b * HID + tid];
    sa[(size_t)b * 128 + tid] = gp > 0.f ? gp : 0.f;
  } else if (tid < 2 * HID) {
    int h = tid - HID;
    float acc = sh_sc[0] * Ub[h];
    const float* urow = Uw + h * LD;
    for (int d = 0; d < LD; ++d) acc += sh_mm[d] * urow[d];
    float lp = acc * invS;
    sa[(size_t)b * 128 + tid] = lp > 0.f ? lp : 0.f;
  } else if (tid < 128) {
    sa[(size_t)b * 128 + tid] = actions[(size_t)b * AD + (tid - 64)];
  }
}

// out[M=4096, N=256] = act(in[M,K] @ w[N,K].T + bias) via fp32 WMMA. 8 waves: 32 rows x 256 cols per block.
__global__ void mlp_kernel(const float* __restrict__ in, const float* __restrict__ w,
                           const float* __restrict__ bias, float* __restrict__ out,
                           int K, int do_relu) {
  int tid = threadIdx.x;
  int lane = tid & 31, wave = tid >> 5;
  int r0 = blockIdx.x * 32 + (wave >> 2) * 16;
  int n0 = (wave & 3) * 64;
  int m = lane & 15, half = lane >> 4;
  v8f acc[4] = {{}, {}, {}, {}};
  const float* arow = in + (size_t)(r0 + m) * K;
  for (int kb = 0; kb < K; kb += 4) {
    int k = kb + 2 * half;
    v2f a; a.x = arow[k]; a.y = arow[k + 1];
#pragma unroll
    for (int t = 0; t < 4; ++t) {
      const float* brow = w + (size_t)(n0 + t * 16 + m) * K;
      v2f bv; bv.x = brow[k]; bv.y = brow[k + 1];
      acc[t] = wmma_f32(a, bv, acc[t]);
    }
  }
#pragma unroll
  for (int t = 0; t < 4; ++t) {
    int n = n0 + t * 16 + m;
    float bs = bias[n];
#pragma unroll
    for (int g = 0; g < 8; ++g) {
      int row = r0 + g + 8 * half;
      float x = acc[t][g] + bs;
      if (do_relu) x = x > 0.f ? x : 0.f;
      out[(size_t)row * 256 + n] = x;
    }
  }
}

// q[b] = h[b,:] . w + bias   (N=1 head). One wave per row.
__global__ void head_kernel(const float* __restrict__ h, const float* __restrict__ w,
                            const float* __restrict__ bias, float* __restrict__ out) {
  int tid = threadIdx.x;
  int lane = tid & 31, wave = tid >> 5;
  int b = blockIdx.x * 8 + wave;
  const float4* hb = (const float4*)(h + (size_t)b * 256);
  const float4* wb = (const float4*)w;
  float d = 0.f;
#pragma unroll
  for (int i = 0; i < 2; ++i) {
    float4 x = hb[lane * 2 + i];
    float4 ww = wb[lane * 2 + i];
    d += x.x * ww.x + x.y * ww.y + x.z * ww.z + x.w * ww.w;
  }
#pragma unroll
  for (int off = 16; off; off >>= 1) d += __shfl_xor(d, off);
  if (lane == 0) out[b] = d + bias[0];
}

extern "C" void kernel_launch(void* const* d_in, const int* in_sizes, int n_in,
                              void* d_out, int out_size, void* d_ws, size_t ws_size,
                              hipStream_t stream) {
  const float* gs   = (const float*)d_in[0];
  const float* ls   = (const float*)d_in[1];
  const float* act  = (const float*)d_in[2];
  const float* Ww   = (const float*)d_in[3];
  const float* Wb   = (const float*)d_in[4];
  const float* Uw   = (const float*)d_in[5];
  const float* Ub   = (const float*)d_in[6];
  const float* attw = (const float*)d_in[7];
  const float* attb = (const float*)d_in[8];
  const float* l1w  = (const float*)d_in[9];
  const float* l1b  = (const float*)d_in[10];
  const float* l2w  = (const float*)d_in[11];
  const float* l2b  = (const float*)d_in[12];
  const float* l3w  = (const float*)d_in[13];
  const float* l3b  = (const float*)d_in[14];
  const float* l4w  = (const float*)d_in[15];
  const float* l4b  = (const float*)d_in[16];
  const float* l5w  = (const float*)d_in[17];
  const float* l5b  = (const float*)d_in[18];
  const float* l6w  = (const float*)d_in[19];
  const float* l6b  = (const float*)d_in[20];

  float* ws = (float*)d_ws;
  float* v  = ws;                  // 128
  float* k0 = ws + 128;            // 1 (padded)
  float* wg = ws + 256;            // 4096*32
  float* cb = wg + B_SZ * HID;     // 4096
  float* s0 = cb + B_SZ;           // 4096
  float* sa = s0 + B_SZ;           // 4096*128
  float* h1 = sa + (size_t)B_SZ * 128;  // 4096*256
  float* h2 = h1 + (size_t)B_SZ * 256;  // 4096*256
  float* q1 = (float*)d_out;
  float* q2 = q1 + B_SZ;

  prep_kernel<<<1, 128, 0, stream>>>(attw, attb, Uw, Ub, v, k0);
  wg_kernel<<<64, 128, 0, stream>>>(gs, Ww, Wb, wg);
  sconst_kernel<<<16, 256, 0, stream>>>(wg, attw, attb, k0, cb, s0);
  attn_kernel<<<B_SZ, 256, 0, stream>>>(ls, v, cb, s0, wg, Uw, Ub, act, sa);
  // tower 1
  mlp_kernel<<<128, 256, 0, stream>>>(sa, l1w, l1b, h1, 128, 1);
  mlp_kernel<<<128, 256, 0, stream>>>(h1, l2w, l2b, h2, 256, 1);
  head_kernel<<<512, 256, 0, stream>>>(h2, l3w, l3b, q1);
  // tower 2
  mlp_kernel<<<128, 256, 0, stream>>>(sa, l4w, l4b, h1, 128, 1);
  mlp_kernel<<<128, 256, 0, stream>>>(h1, l5w, l5b, h2, 256, 1);
  head_kernel<<<512, 256, 0, stream>>>(h2, l6w, l6b, q2);
}